// MultiSimilarityLoss_72481868087461
// MI455X (gfx1250) — compile-verified
//
#include <hip/hip_runtime.h>

#define B_N 4096
#define D_K 128
#define NCOLT (B_N / 16)   // 256 column tiles
#define WAVES 4            // waves per block (wave32 -> 128 threads)

#define ALPHA_C 1.0f
#define BETA_C 10.0f
#define MARGIN_C 0.5f
#define EPS_C 0.1f

typedef __attribute__((ext_vector_type(2))) float v2f;
typedef __attribute__((ext_vector_type(8))) float v8f;

// D = A(16x4 f32) * B(4x16 f32) + C(16x16 f32), wave32 WMMA
__device__ __forceinline__ v8f wmma_k4(v2f a, v2f b, v8f c) {
  return __builtin_amdgcn_wmma_f32_16x16x4_f32(
      /*neg_a=*/false, a, /*neg_b=*/false, b,
      /*c_mod=*/(short)0, c, /*reuse_a=*/false, /*reuse_b=*/false);
}

// -------- Pass 1: per-row hardest mining: negMax[i], posMin[i] --------
__global__ __launch_bounds__(WAVES * 32) void ms_mine_kernel(
    const float* __restrict__ E, const int* __restrict__ labels,
    float* __restrict__ negMax, float* __restrict__ posMin) {
  const int rowBase = blockIdx.x * 16;
  const int tid  = threadIdx.x;
  const int wave = tid >> 5;
  const int lane = tid & 31;
  const int half = lane >> 4;   // which 16-lane half of the wave
  const int idx  = lane & 15;

  // Preload A fragments: A[m][k] = E[rowBase+m][k].
  // 16x4 f32 A layout: lanes 0-15 hold M=idx with K = kBase+{0,1};
  // lanes 16-31 hold M=idx with K = kBase+{2,3}.
  const float* arow = E + (rowBase + idx) * D_K;
  v2f afrag[32];
#pragma unroll
  for (int t = 0; t < 32; ++t) {
    const float2 av = *(const float2*)(arow + 4 * t + 2 * half);
    afrag[t].x = av.x; afrag[t].y = av.y;
  }

  int lrow[8];
#pragma unroll
  for (int r = 0; r < 8; ++r) lrow[r] = labels[rowBase + r + 8 * half];

  float rmax[8], rmin[8];
#pragma unroll
  for (int r = 0; r < 8; ++r) { rmax[r] = -__builtin_inff(); rmin[r] = __builtin_inff(); }

  for (int ct = wave; ct < NCOLT; ct += WAVES) {
    const int colBase = ct * 16;
    const float* brow = E + (colBase + idx) * D_K;  // B[k][n] = E[colBase+n][k]
    v8f c = {0.f, 0.f, 0.f, 0.f, 0.f, 0.f, 0.f, 0.f};
#pragma unroll
    for (int t = 0; t < 32; ++t) {
      const float2 bv = *(const float2*)(brow + 4 * t + 2 * half);
      v2f b; b.x = bv.x; b.y = bv.y;
      c = wmma_k4(afrag[t], b, c);
    }
    const int col  = colBase + idx;
    const int lcol = labels[col];
#pragma unroll
    for (int r = 0; r < 8; ++r) {
      const int row = rowBase + r + 8 * half;   // C layout: VGPR r, half -> M
      const float s = c[r];
      const bool offd = (row != col);
      const bool eq = (lrow[r] == lcol);
      if (offd && eq)  rmin[r] = fminf(rmin[r], s);
      if (offd && !eq) rmax[r] = fmaxf(rmax[r], s);
    }
  }

  // butterfly reduce within each 16-lane half (row lives in one half)
#pragma unroll
  for (int r = 0; r < 8; ++r) {
#pragma unroll
    for (int m = 8; m >= 1; m >>= 1) {
      rmin[r] = fminf(rmin[r], __shfl_xor(rmin[r], m, 32));
      rmax[r] = fmaxf(rmax[r], __shfl_xor(rmax[r], m, 32));
    }
  }

  __shared__ float sMin[WAVES][16];
  __shared__ float sMax[WAVES][16];
  if (idx == 0) {
#pragma unroll
    for (int r = 0; r < 8; ++r) {
      sMin[wave][r + 8 * half] = rmin[r];
      sMax[wave][r + 8 * half] = rmax[r];
    }
  }
  __syncthreads();
  if (tid < 16) {
    float mn = sMin[0][tid], mx = sMax[0][tid];
#pragma unroll
    for (int w = 1; w < WAVES; ++w) {
      mn = fminf(mn, sMin[w][tid]);
      mx = fmaxf(mx, sMax[w][tid]);
    }
    posMin[rowBase + tid] = mn;
    negMax[rowBase + tid] = mx;
  }
}

// -------- Pass 2: hard-example exp sums: posSum[i], negSum[i] --------
__global__ __launch_bounds__(WAVES * 32) void ms_sum_kernel(
    const float* __restrict__ E, const int* __restrict__ labels,
    const float* __restrict__ negMax, const float* __restrict__ posMin,
    float* __restrict__ posSum, float* __restrict__ negSum) {
  const int rowBase = blockIdx.x * 16;
  const int tid  = threadIdx.x;
  const int wave = tid >> 5;
  const int lane = tid & 31;
  const int half = lane >> 4;
  const int idx  = lane & 15;

  const float* arow = E + (rowBase + idx) * D_K;
  v2f afrag[32];
#pragma unroll
  for (int t = 0; t < 32; ++t) {
    const float2 av = *(const float2*)(arow + 4 * t + 2 * half);
    afrag[t].x = av.x; afrag[t].y = av.y;
  }

  int lrow[8];
  float thP[8], thN[8];
#pragma unroll
  for (int r = 0; r < 8; ++r) {
    const int row = rowBase + r + 8 * half;
    lrow[r] = labels[row];
    thP[r] = negMax[row] + EPS_C;   // hard_pos: sim <  negMax + eps
    thN[r] = posMin[row] - EPS_C;   // hard_neg: sim >  posMin - eps
  }

  float ps[8], ns[8];
#pragma unroll
  for (int r = 0; r < 8; ++r) { ps[r] = 0.f; ns[r] = 0.f; }

  for (int ct = wave; ct < NCOLT; ct += WAVES) {
    const int colBase = ct * 16;
    const float* brow = E + (colBase + idx) * D_K;
    v8f c = {0.f, 0.f, 0.f, 0.f, 0.f, 0.f, 0.f, 0.f};
#pragma unroll
    for (int t = 0; t < 32; ++t) {
      const float2 bv = *(const float2*)(brow + 4 * t + 2 * half);
      v2f b; b.x = bv.x; b.y = bv.y;
      c = wmma_k4(afrag[t], b, c);
    }
    const int col  = colBase + idx;
    const int lcol = labels[col];
#pragma unroll
    for (int r = 0; r < 8; ++r) {
      const int row = rowBase + r + 8 * half;
      const float s = c[r];
      const bool offd = (row != col);
      const bool eq = (lrow[r] == lcol);
      if (offd && eq  && s < thP[r]) ps[r] += expf(-ALPHA_C * (s - MARGIN_C));
      if (offd && !eq && s > thN[r]) ns[r] += expf( BETA_C  * (s - MARGIN_C));
    }
  }

#pragma unroll
  for (int r = 0; r < 8; ++r) {
#pragma unroll
    for (int m = 8; m >= 1; m >>= 1) {
      ps[r] += __shfl_xor(ps[r], m, 32);
      ns[r] += __shfl_xor(ns[r], m, 32);
    }
  }

  __shared__ float sP[WAVES][16];
  __shared__ float sN[WAVES][16];
  if (idx == 0) {
#pragma unroll
    for (int r = 0; r < 8; ++r) {
      sP[wave][r + 8 * half] = ps[r];
      sN[wave][r + 8 * half] = ns[r];
    }
  }
  __syncthreads();
  if (tid < 16) {
    float p = sP[0][tid], n = sN[0][tid];
#pragma unroll
    for (int w = 1; w < WAVES; ++w) { p += sP[w][tid]; n += sN[w][tid]; }
    posSum[rowBase + tid] = p;
    negSum[rowBase + tid] = n;
  }
}

// -------- Pass 3: per-anchor terms, validity, mean --------
__global__ __launch_bounds__(256) void ms_final_kernel(
    const float* __restrict__ posSum, const float* __restrict__ negSum,
    float* __restrict__ out) {
  __shared__ float sT[256];
  __shared__ int   sC[256];
  const int tid = threadIdx.x;
  float tot = 0.f;
  int cnt = 0;
  for (int i = tid; i < B_N; i += 256) {
    const float p = posSum[i];
    const float n = negSum[i];
    // exp terms are strictly positive, and hard_* nonempty implies mask
    // nonempty, so (p>0 && n>0) == reference's full validity condition.
    if (p > 0.f && n > 0.f) {
      tot += (1.0f / ALPHA_C) * log1pf(p) + (1.0f / BETA_C) * log1pf(n);
      cnt += 1;
    }
  }
  sT[tid] = tot;
  sC[tid] = cnt;
  __syncthreads();
  for (int s = 128; s > 0; s >>= 1) {
    if (tid < s) { sT[tid] += sT[tid + s]; sC[tid] += sC[tid + s]; }
    __syncthreads();
  }
  if (tid == 0) {
    const int nv = sC[0] > 1 ? sC[0] : 1;
    out[0] = sT[0] / (float)nv;
  }
}

extern "C" void kernel_launch(void* const* d_in, const int* in_sizes, int n_in,
                              void* d_out, int out_size, void* d_ws, size_t ws_size,
                              hipStream_t stream) {
  (void)in_sizes; (void)n_in; (void)out_size; (void)ws_size;
  const float* E      = (const float*)d_in[0];
  const int*   labels = (const int*)d_in[1];
  float* out = (float*)d_out;

  float* negMax = (float*)d_ws;          // [4096]
  float* posMin = negMax + B_N;          // [4096]
  float* posSum = posMin + B_N;          // [4096]
  float* negSum = posSum + B_N;          // [4096]  total ws use: 64 KB

  ms_mine_kernel<<<dim3(NCOLT), dim3(WAVES * 32), 0, stream>>>(E, labels, negMax, posMin);
  ms_sum_kernel<<<dim3(NCOLT), dim3(WAVES * 32), 0, stream>>>(E, labels, negMax, posMin,
                                                              posSum, negSum);
  ms_final_kernel<<<1, 256, 0, stream>>>(posSum, negSum, out);
}